// Seg_1451698946593
// MI455X (gfx1250) — compile-verified
//
#include <hip/hip_runtime.h>
#include <hip/hip_bf16.h>
#include <math.h>

typedef __attribute__((ext_vector_type(2))) float v2f;
typedef __attribute__((ext_vector_type(8))) float v8f;

#define BATCH 4
#define NPTS  4096
#define EPS_BN  1e-5f
#define EPS_ATT 1e-9f

// ---------------- BN folding: y = s*acc + t  (t absorbs conv bias) -------
__global__ __launch_bounds__(64) void k_fold_bn(const float* __restrict__ g,
                                                const float* __restrict__ bb,
                                                const float* __restrict__ m,
                                                const float* __restrict__ v,
                                                const float* __restrict__ bias,
                                                float* __restrict__ s,
                                                float* __restrict__ t, int C) {
  int c = blockIdx.x * 64 + threadIdx.x;
  if (c >= C) return;
  float sv = g[c] * rsqrtf(v[c] + EPS_BN);
  float tv = bb[c] - sv * m[c];
  if (bias) tv += sv * bias[c];
  s[c] = sv; t[c] = tv;
}

// ---------------- conv1 (Cin=6) + bn + relu ------------------------------
__global__ __launch_bounds__(256) void k_conv1(const float* __restrict__ x,
                                               const float* __restrict__ W1,
                                               const float* __restrict__ s,
                                               const float* __restrict__ t,
                                               float* __restrict__ h1) {
  int n = blockIdx.x * 256 + threadIdx.x;
  int c = blockIdx.y, b = blockIdx.z;
  float acc = 0.f;
#pragma unroll
  for (int i = 0; i < 6; ++i)
    acc += W1[c * 6 + i] * x[((long)b * 6 + i) * NPTS + n];
  float y = s[c] * acc + t[c];
  h1[((long)b * 128 + c) * NPTS + n] = fmaxf(y, 0.f);
}

// ---------------- generic WMMA GEMM: 16(cout) x 64(n) strip per wave -----
// Y[b,o,n] = act(s*( W@X + colAdd ) + t) + res
// W row-major [Cout, ldw] (uses first Cin columns); X [b, Cin, NPTS]
__global__ __launch_bounds__(128) void k_gemm(
    const float* __restrict__ W, int ldw, int Cin, int Cout,
    const float* __restrict__ X, long xBS,
    const float* __restrict__ s, const float* __restrict__ t,
    const float* __restrict__ colAdd,
    const float* __restrict__ res, long resBS,
    float* __restrict__ Y, long yBS, int act) {
  const int wave = threadIdx.x >> 5;
  const int lane = threadIdx.x & 31;
  const int ln = lane & 15, hi = lane >> 4;
  const int n0 = (blockIdx.x * 4 + wave) * 64;
  const int m0 = blockIdx.y * 16;
  const int b  = blockIdx.z;
  const float* Wr = W + (long)(m0 + ln) * ldw + 2 * hi;      // A: K pair {0,1}/{2,3}
  const float* Xc = X + (long)b * xBS + (long)(2 * hi) * NPTS + n0 + ln;
  v8f acc[4] = {};
  for (int k = 0; k < Cin; k += 4) {
    __builtin_prefetch(Xc + (long)(k + 8) * NPTS, 0, 0);
    v2f a = *(const v2f*)(Wr + k);
    const long o0 = (long)k * NPTS, o1 = o0 + NPTS;
#pragma unroll
    for (int j = 0; j < 4; ++j) {
      v2f bv2;
      bv2.x = Xc[o0 + 16 * j];
      bv2.y = Xc[o1 + 16 * j];
      acc[j] = __builtin_amdgcn_wmma_f32_16x16x4_f32(false, a, false, bv2,
                                                     (short)0, acc[j], false, false);
    }
  }
#pragma unroll
  for (int j = 0; j < 4; ++j) {
    const int n = n0 + 16 * j + ln;
#pragma unroll
    for (int r = 0; r < 8; ++r) {
      int o = m0 + r + 8 * hi;
      float y = acc[j][r];
      if (colAdd) y += colAdd[b * Cout + o];
      if (s)      y = s[o] * y + t[o];
      else if (t) y += t[o];
      if (act == 1)      y = fmaxf(y, 0.f);
      else if (act == 2) y = (y > 0.f) ? y : 0.2f * y;
      if (res) y += res[(long)b * resBS + (long)o * NPTS + n];
      Y[(long)b * yBS + (long)o * NPTS + n] = y;
    }
  }
}

// ---------------- energy = Qk^T @ Qk, Q [b,32,N], E [b,N,N] --------------
// 16(row) x 64(col) strip per wave
__global__ __launch_bounds__(128) void k_energy(const float* __restrict__ Q,
                                                float* __restrict__ E) {
  const int wave = threadIdx.x >> 5;
  const int lane = threadIdx.x & 31;
  const int ln = lane & 15, hi = lane >> 4;
  const int c0 = (blockIdx.x * 4 + wave) * 64;  // E column (m)
  const int r0 = blockIdx.y * 16;               // E row (n)
  const int b  = blockIdx.z;
  const float* Qb = Q + (long)b * 32 * NPTS + (long)(2 * hi) * NPTS;
  const float* Qa = Qb + r0 + ln;
  const float* Qc = Qb + c0 + ln;
  v8f acc[4] = {};
#pragma unroll
  for (int k = 0; k < 32; k += 4) {
    v2f a;
    a.x = Qa[(long)k * NPTS];
    a.y = Qa[(long)(k + 1) * NPTS];
    const long o0 = (long)k * NPTS, o1 = o0 + NPTS;
#pragma unroll
    for (int j = 0; j < 4; ++j) {
      v2f bv2;
      bv2.x = Qc[o0 + 16 * j];
      bv2.y = Qc[o1 + 16 * j];
      acc[j] = __builtin_amdgcn_wmma_f32_16x16x4_f32(false, a, false, bv2,
                                                     (short)0, acc[j], false, false);
    }
  }
  float* Eb = E + (long)b * NPTS * NPTS;
#pragma unroll
  for (int j = 0; j < 4; ++j)
#pragma unroll
    for (int r = 0; r < 8; ++r)
      Eb[(long)(r0 + r + 8 * hi) * NPTS + c0 + 16 * j + ln] = acc[j][r];
}

// ---------------- row softmax, in place ----------------------------------
__global__ __launch_bounds__(256) void k_softmax(float* __restrict__ E) {
  const long row = blockIdx.x;            // b*N + n
  float* p = E + row * (long)NPTS;
  __shared__ float red[8];
  const int tid = threadIdx.x, lane = tid & 31, wid = tid >> 5;
  float vals[16];
  float lmax = -3.4e38f;
#pragma unroll
  for (int i = 0; i < 16; ++i) { vals[i] = p[tid + i * 256]; lmax = fmaxf(lmax, vals[i]); }
#pragma unroll
  for (int off = 16; off; off >>= 1) lmax = fmaxf(lmax, __shfl_xor(lmax, off, 32));
  if (lane == 0) red[wid] = lmax;
  __syncthreads();
  float rmax = red[0];
#pragma unroll
  for (int i = 1; i < 8; ++i) rmax = fmaxf(rmax, red[i]);
  __syncthreads();
  float lsum = 0.f;
#pragma unroll
  for (int i = 0; i < 16; ++i) { vals[i] = __expf(vals[i] - rmax); lsum += vals[i]; }
#pragma unroll
  for (int off = 16; off; off >>= 1) lsum += __shfl_xor(lsum, off, 32);
  if (lane == 0) red[wid] = lsum;
  __syncthreads();
  float rsum = 0.f;
#pragma unroll
  for (int i = 0; i < 8; ++i) rsum += red[i];
  float rinv = 1.f / rsum;
#pragma unroll
  for (int i = 0; i < 16; ++i) p[tid + i * 256] = vals[i] * rinv;
}

// ---------------- deterministic column sums ------------------------------
__global__ __launch_bounds__(256) void k_colsum(const float* __restrict__ E,
                                                float* __restrict__ cs) {
  int m = blockIdx.x * 256 + threadIdx.x;
  int b = blockIdx.y;
  const float* p = E + (long)b * NPTS * NPTS + m;
  float acc = 0.f;
  for (int n = 0; n < NPTS; ++n) acc += p[(long)n * NPTS];
  cs[b * NPTS + m] = acc;
}

// ---------------- u = x - (xv @ attn) / (eps + colsum) -------------------
// 16(chan) x 64(point) strip per wave; K = NPTS
__global__ __launch_bounds__(128) void k_attn_apply(
    const float* __restrict__ XV, const float* __restrict__ E,
    const float* __restrict__ cs,
    const float* __restrict__ Xin, long xinBS,
    float* __restrict__ U) {
  const int wave = threadIdx.x >> 5;
  const int lane = threadIdx.x & 31;
  const int ln = lane & 15, hi = lane >> 4;
  const int m0 = (blockIdx.x * 4 + wave) * 64;  // point tile base
  const int c0 = blockIdx.y * 16;               // channel tile
  const int b  = blockIdx.z;
  const float* Ar = XV + ((long)b * 128 + c0 + ln) * NPTS + 2 * hi;  // contiguous K pair
  const float* Bc = E + (long)b * NPTS * NPTS + (long)(2 * hi) * NPTS + m0 + ln;
  v8f acc[4] = {};
  for (int k = 0; k < NPTS; k += 4) {
    __builtin_prefetch(Bc + (long)(k + 16) * NPTS, 0, 0);
    v2f a = *(const v2f*)(Ar + k);
    const long o0 = (long)k * NPTS, o1 = o0 + NPTS;
#pragma unroll
    for (int j = 0; j < 4; ++j) {
      v2f bv2;
      bv2.x = Bc[o0 + 16 * j];
      bv2.y = Bc[o1 + 16 * j];
      acc[j] = __builtin_amdgcn_wmma_f32_16x16x4_f32(false, a, false, bv2,
                                                     (short)0, acc[j], false, false);
    }
  }
#pragma unroll
  for (int j = 0; j < 4; ++j) {
    const int m = m0 + 16 * j + ln;
    const float rcs = 1.f / (EPS_ATT + cs[b * NPTS + m]);
#pragma unroll
    for (int r = 0; r < 8; ++r) {
      int c = c0 + r + 8 * hi;
      U[((long)b * 128 + c) * NPTS + m] =
          Xin[(long)b * xinBS + (long)c * NPTS + m] - acc[j][r] * rcs;
    }
  }
}

// ---------------- max+avg pool over N ------------------------------------
__global__ __launch_bounds__(256) void k_pool(const float* __restrict__ xf,
                                              float* __restrict__ glob) {
  int c = blockIdx.x, b = blockIdx.y;
  const float* p = xf + ((long)b * 1024 + c) * NPTS;
  const int tid = threadIdx.x, lane = tid & 31, wid = tid >> 5;
  __shared__ float rmx[8], rsm[8];
  float mx = -3.4e38f, sm = 0.f;
  for (int i = tid; i < NPTS; i += 256) { float v = p[i]; mx = fmaxf(mx, v); sm += v; }
#pragma unroll
  for (int off = 16; off; off >>= 1) { mx = fmaxf(mx, __shfl_xor(mx, off, 32)); sm += __shfl_xor(sm, off, 32); }
  if (lane == 0) { rmx[wid] = mx; rsm[wid] = sm; }
  __syncthreads();
  if (tid == 0) {
    float gmx = rmx[0], gsm = rsm[0];
    for (int i = 1; i < 8; ++i) { gmx = fmaxf(gmx, rmx[i]); gsm += rsm[i]; }
    glob[(long)b * 2048 + c] = gmx;
    glob[(long)b * 2048 + 1024 + c] = gsm * (1.f / NPTS);
  }
}

// -------- g1[b,o] = bs1[o] + Ws1[o,1024:3072] @ glob[b]  (rank-1 branch) --
__global__ __launch_bounds__(256) void k_globproj(const float* __restrict__ Ws1,
                                                  const float* __restrict__ glob,
                                                  const float* __restrict__ bs1,
                                                  float* __restrict__ g1) {
  int idx = blockIdx.x * 256 + threadIdx.x;
  if (idx >= BATCH * 512) return;
  int b = idx >> 9, o = idx & 511;
  const float* w = Ws1 + (long)o * 3072 + 1024;
  const float* g = glob + (long)b * 2048;
  float acc = bs1[o];
  for (int c = 0; c < 2048; ++c) acc += w[c] * g[c];
  g1[idx] = acc;
}

// ---------------- final 13-channel conv ----------------------------------
__global__ __launch_bounds__(256) void k_final(const float* __restrict__ hs2,
                                               const float* __restrict__ Ws3,
                                               const float* __restrict__ bs3,
                                               float* __restrict__ out) {
  int n = blockIdx.x * 256 + threadIdx.x;
  int o = blockIdx.y, b = blockIdx.z;
  const float* w = Ws3 + o * 256;
  const float* h = hs2 + (long)b * 256 * NPTS + n;
  float acc = bs3[o];
  for (int c = 0; c < 256; ++c) acc += w[c] * h[(long)c * NPTS];
  out[((long)b * 13 + o) * NPTS + n] = acc;
}

extern "C" void kernel_launch(void* const* d_in, const int* in_sizes, int n_in,
                              void* d_out, int out_size, void* d_ws, size_t ws_size,
                              hipStream_t stream) {
  (void)in_sizes; (void)n_in; (void)out_size; (void)ws_size;
  auto in = [&](int i) { return (const float*)d_in[i]; };
  float* ws = (float*)d_ws;
  const long NC = (long)BATCH * NPTS;
  long off = 0;
  float* h1  = ws + off; off += NC * 128;
  float* hb  = ws + off; off += NC * 128;
  float* xc  = ws + off; off += NC * 512;   // concat of x1..x4
  float* xqk = ws + off; off += NC * 32;
  float* xv  = ws + off; off += NC * 128;
  float* u   = ws + off; off += NC * 128;
  float* xf  = ws + off; off += NC * 1024;
  float* hs1 = ws + off; off += NC * 512;
  float* hs2 = ws + off; off += NC * 256;
  float* glob = ws + off; off += BATCH * 2048;
  float* g1   = ws + off; off += BATCH * 512;
  float* cs   = ws + off; off += BATCH * NPTS;
  float* st   = ws + off; off += 16384;
  float* E    = ws + off; off += (long)BATCH * NPTS * NPTS;

  float *s1 = st,        *t1 = st + 128;
  float *s2 = st + 256,  *t2 = st + 384;
  float *ssa[4], *tsa[4];
  for (int i = 0; i < 4; ++i) { ssa[i] = st + 512 + i * 256; tsa[i] = ssa[i] + 128; }
  float *sf  = st + 1536, *tf  = st + 2560;
  float *ss1 = st + 3584, *ts1 = st + 4096;
  float *ss2 = st + 4608, *ts2 = st + 4864;

  // ---- fold all BN params ----
  k_fold_bn<<<2, 64, 0, stream>>>(in(2), in(3), in(4), in(5), nullptr, s1, t1, 128);
  k_fold_bn<<<2, 64, 0, stream>>>(in(7), in(8), in(9), in(10), nullptr, s2, t2, 128);
  for (int i = 0; i < 4; ++i) {
    int b0 = 11 + 9 * i;  // Wqk,Wv,bv,Wt,bt,bn(g,b,m,v)
    k_fold_bn<<<2, 64, 0, stream>>>(in(b0 + 5), in(b0 + 6), in(b0 + 7), in(b0 + 8),
                                    in(b0 + 4), ssa[i], tsa[i], 128);
  }
  k_fold_bn<<<16, 64, 0, stream>>>(in(48), in(49), in(50), in(51), nullptr, sf, tf, 1024);
  k_fold_bn<<<8, 64, 0, stream>>>(in(54), in(55), in(56), in(57), nullptr, ss1, ts1, 512);
  k_fold_bn<<<4, 64, 0, stream>>>(in(60), in(61), in(62), in(63), in(59), ss2, ts2, 256);

  const dim3 blk(128);
  const long C128 = (long)128 * NPTS, C512 = (long)512 * NPTS;
  const int GX = NPTS / 256;  // 16 : x-grid for 64-wide wave strips, 4 waves/block

  // ---- stem ----
  k_conv1<<<dim3(NPTS / 256, 128, BATCH), 256, 0, stream>>>(in(0), in(1), s1, t1, h1);
  k_gemm<<<dim3(GX, 8, BATCH), blk, 0, stream>>>(in(6), 128, 128, 128, h1, C128,
        s2, t2, nullptr, nullptr, 0, hb, C128, 1);

  // ---- 4 offset-attention layers, outputs written into concat slots ----
  for (int i = 0; i < 4; ++i) {
    int b0 = 11 + 9 * i;
    const float* xin = (i == 0) ? hb : xc + (long)(i - 1) * 128 * NPTS;
    long xinBS = (i == 0) ? C128 : C512;
    k_gemm<<<dim3(GX, 2, BATCH), blk, 0, stream>>>(in(b0), 128, 128, 32, xin, xinBS,
          nullptr, nullptr, nullptr, nullptr, 0, xqk, (long)32 * NPTS, 0);
    k_gemm<<<dim3(GX, 8, BATCH), blk, 0, stream>>>(in(b0 + 1), 128, 128, 128, xin, xinBS,
          nullptr, in(b0 + 2), nullptr, nullptr, 0, xv, C128, 0);
    k_energy<<<dim3(GX, NPTS / 16, BATCH), blk, 0, stream>>>(xqk, E);
    k_softmax<<<dim3(BATCH * NPTS), 256, 0, stream>>>(E);
    k_colsum<<<dim3(NPTS / 256, BATCH), 256, 0, stream>>>(E, cs);
    k_attn_apply<<<dim3(GX, 8, BATCH), blk, 0, stream>>>(xv, E, cs, xin, xinBS, u);
    k_gemm<<<dim3(GX, 8, BATCH), blk, 0, stream>>>(in(b0 + 3), 128, 128, 128, u, C128,
          ssa[i], tsa[i], nullptr, xin, xinBS, xc + (long)i * 128 * NPTS, C512, 1);
  }

  // ---- head ----
  k_gemm<<<dim3(GX, 64, BATCH), blk, 0, stream>>>(in(47), 512, 512, 1024, xc, C512,
        sf, tf, nullptr, nullptr, 0, xf, (long)1024 * NPTS, 2);
  k_pool<<<dim3(1024, BATCH), 256, 0, stream>>>(xf, glob);
  k_globproj<<<dim3((BATCH * 512 + 255) / 256), 256, 0, stream>>>(in(52), glob, in(53), g1);
  k_gemm<<<dim3(GX, 32, BATCH), blk, 0, stream>>>(in(52), 3072, 1024, 512, xf,
        (long)1024 * NPTS, ss1, ts1, g1, nullptr, 0, hs1, C512, 1);
  k_gemm<<<dim3(GX, 16, BATCH), blk, 0, stream>>>(in(58), 512, 512, 256, hs1, C512,
        ss2, ts2, nullptr, nullptr, 0, hs2, (long)256 * NPTS, 1);
  k_final<<<dim3(NPTS / 256, 13, BATCH), 256, 0, stream>>>(hs2, in(64), in(65), (float*)d_out);
}